// BranchMarkovLayer_3221225472541
// MI455X (gfx1250) — compile-verified
//
#include <hip/hip_runtime.h>
#include <math.h>

typedef __attribute__((ext_vector_type(2))) float v2f;
typedef __attribute__((ext_vector_type(8))) float v8f;
typedef int b128_t __attribute__((vector_size(16)));

#define BTOT   131072
#define COLS   192      // X row = 3*64 floats
#define SAMPS  32       // samples per block (2 waves x 16)
#define TPB    64
#define NCH    8        // node chunks (8 nodes each)
#define ZSTR   68       // 64 data + 4 pad floats (bank spread, 16B-aligned rows)
#define XSTR   76       // 64 own + 8 par + 1 root + pad
#define YSTR   68
#define HSTR   20       // 16 h-cols + pad (16B-aligned rows)

#if __has_builtin(__builtin_amdgcn_global_load_async_to_lds_b128)
#define HAVE_ASYNC 1
#else
#define HAVE_ASYNC 0
#endif

#if HAVE_ASYNC
__device__ __forceinline__ void async_copy16(const float* g, float* l) {
  __builtin_amdgcn_global_load_async_to_lds_b128(
      (__attribute__((address_space(1))) b128_t*)(g),
      (__attribute__((address_space(3))) b128_t*)(l),
      0, 0);
}
#endif

__device__ __forceinline__ float fast_rcp(float x) {
#if __has_builtin(__builtin_amdgcn_rcpf)
  return __builtin_amdgcn_rcpf(x);
#else
  return 1.0f / x;
#endif
}

// ---------------- pass 1: per-column partial sums of log1p(x) ----------------
__global__ void k_stats_partial(const float* __restrict__ X,
                                float* __restrict__ part,  // [nb][2][COLS]
                                int nb, int chunk) {
  const int j   = threadIdx.x;     // column 0..191
  const int blk = blockIdx.x;
  long r0 = (long)blk * chunk;
  long r1 = r0 + chunk; if (r1 > BTOT) r1 = BTOT;
  double s = 0.0, s2 = 0.0;
  for (long r = r0; r < r1; ++r) {
    float v  = X[r * COLS + j];
    float lv = logf(v + 1.0f);
    s  += (double)lv;
    s2 += (double)lv * (double)lv;
  }
  part[(long)blk * (2 * COLS) + j]        = (float)s;
  part[(long)blk * (2 * COLS) + COLS + j] = (float)s2;
}

// ------------- pass 2: deterministic reduce -> mu, inv_sd (ddof=1) ----------
__global__ void k_stats_final(float* __restrict__ ws, int nb) {
  const int j = threadIdx.x;
  const float* part = ws + 2 * COLS;
  double S = 0.0, S2 = 0.0;
  for (int b = 0; b < nb; ++b) {
    S  += (double)part[(long)b * (2 * COLS) + j];
    S2 += (double)part[(long)b * (2 * COLS) + COLS + j];
  }
  double mu  = S / (double)BTOT;
  double var = (S2 - (double)BTOT * mu * mu) / (double)(BTOT - 1);
  if (var < 1e-30) var = 1e-30;
  ws[j]        = (float)mu;
  ws[COLS + j] = (float)(1.0 / sqrt(var));
}

// ---------------- pass 3: fused normalize + per-node MLP via WMMA -----------
__global__ __launch_bounds__(TPB)
void k_mlp(const float* __restrict__ X,  const float* __restrict__ Z,
           const float* __restrict__ W1, const float* __restrict__ b1,
           const float* __restrict__ W2, const float* __restrict__ b2,
           const float* __restrict__ stats, float* __restrict__ out) {
  __shared__ __align__(16) float Zs[2][SAMPS * ZSTR]; // double-buffered Z tiles
  __shared__ __align__(16) float XF[SAMPS * XSTR];    // normalized own/par/root
  __shared__ __align__(16) float Yl[SAMPS * YSTR];    // staged outputs
  __shared__ __align__(16) float W1L[64 * 66];
  __shared__ __align__(16) float b1P[64 * 16];        // b1 padded to 16 (zero h>=6)
  __shared__ __align__(16) float W2P[64 * 8];         // W2 padded to 8
  __shared__ __align__(16) float b2L[64];
  __shared__ __align__(16) float Hb[2 * 16 * HSTR];   // per-wave h transpose buffer
  __shared__ __align__(16) float muL[COLS];
  __shared__ __align__(16) float isdL[COLS];

  const int tid  = threadIdx.x;
  const int wave = tid >> 5;
  const int lane = tid & 31;
  const long b0  = (long)blockIdx.x * SAMPS;
  const float* Zblk = Z + b0 * 512;

  for (int i = tid; i < COLS; i += TPB) { muL[i] = stats[i]; isdL[i] = stats[COLS + i]; }
  for (int i = tid; i < 64 * 66; i += TPB) W1L[i] = W1[i];
  for (int i = tid; i < 64 * 16; i += TPB) { int n = i >> 4, h = i & 15; b1P[i] = (h < 6) ? b1[n * 6 + h] : 0.0f; }
  for (int i = tid; i < 64 * 8;  i += TPB) { int n = i >> 3, h = i & 7;  W2P[i] = (h < 6) ? W2[n * 6 + h] : 0.0f; }
  for (int i = tid; i < 64;      i += TPB) b2L[i] = b2[i];
  __syncthreads();

  // normalized X features: own (c=2,n), par (c=1,n>>3), root (c=0,0)
  for (int i = tid; i < SAMPS * 64; i += TPB) {
    int s = i >> 6, n = i & 63;
    float v = X[(b0 + s) * COLS + 128 + n];
    XF[s * XSTR + n] = (logf(v + 1.0f) - muL[128 + n]) * isdL[128 + n];
  }
  for (int i = tid; i < SAMPS * 8; i += TPB) {
    int s = i >> 3, p = i & 7;
    float v = X[(b0 + s) * COLS + 64 + p];
    XF[s * XSTR + 64 + p] = (logf(v + 1.0f) - muL[64 + p]) * isdL[64 + p];
  }
  for (int i = tid; i < SAMPS; i += TPB) {
    float v = X[(b0 + i) * COLS];
    XF[i * XSTR + 72] = (logf(v + 1.0f) - muL[0]) * isdL[0];
  }

  const int  m     = lane & 15;
  const bool hi    = lane >= 16;          // lanes 16..31 carry K=2,3 of each chunk
  const int  sw0   = wave * 16;           // this wave's sample base
  const int  fb    = hi ? 2 : 0;
  const int  hc    = (m < 6) ? m : 5;     // clamp hidden index, mask below
  const float wmask = (m < 6) ? 1.0f : 0.0f;
  const float ymask = hi ? 0.0f : wmask;  // K=9 (par) / K=11 (pad) lane mask
  float* hb = &Hb[wave * 16 * HSTR];

  __syncthreads();
  const float rootv = XF[(sw0 + m) * XSTR + 72];  // loop-invariant root feature

  // ---- stage a Z chunk (8 nodes = 64 floats per sample) into buffer `buf` ----
  auto stage = [&](int c, int buf) {
    for (int i = tid; i < SAMPS * 16; i += TPB) {
      int s = i >> 4, q4 = (i & 15) << 2;
#if HAVE_ASYNC
      async_copy16(Zblk + (long)s * 512 + c * 64 + q4, &Zs[buf][s * ZSTR + q4]);
#else
      const float4 zv = *(const float4*)(Zblk + (long)s * 512 + c * 64 + q4);
      *(float4*)(&Zs[buf][s * ZSTR + q4]) = zv;
#endif
    }
  };

  stage(0, 0);

  for (int c = 0; c < NCH; ++c) {
    if (c + 1 < NCH) {
      stage(c + 1, (c + 1) & 1);     // prefetch next tile (prev readers of this
#if HAVE_ASYNC                        // buffer were fenced by last iter's barrier)
#if __has_builtin(__builtin_amdgcn_s_wait_asynccnt)
      __builtin_amdgcn_s_wait_asynccnt(8);   // wait for chunk c only
#else
      asm volatile("s_wait_asynccnt 8" ::: "memory");
#endif
#endif
    } else {
#if HAVE_ASYNC
#if __has_builtin(__builtin_amdgcn_s_wait_asynccnt)
      __builtin_amdgcn_s_wait_asynccnt(0);
#else
      asm volatile("s_wait_asynccnt 0" ::: "memory");
#endif
#endif
    }
    __syncthreads();
    const float* zbuf = Zs[c & 1];

    for (int nn = 0; nn < 8; ++nn) {
      const int n = c * 8 + nn;
      // ---- A operand: data[sample m, feature k] (K chunks of 4) ----
      const float2 za = *(const float2*)(&zbuf[(sw0 + m) * ZSTR + nn * 8 + fb]);      // K 0..3
      const float2 zb = *(const float2*)(&zbuf[(sw0 + m) * ZSTR + nn * 8 + 4 + fb]);  // K 4..7
      float ownv = XF[(sw0 + m) * XSTR + (hi ? 72 : n)];                 // own (root for hi)
      float parv = XF[(sw0 + m) * XSTR + (hi ? 72 : (64 + (n >> 3)))];   // par (dummy for hi)
      v2f A0 = {za.x, za.y};
      v2f A1 = {zb.x, zb.y};
      v2f A2 = {hi ? rootv : ownv, hi ? 0.0f : parv};                    // K 8..11 (11 = pad)
      // ---- B operand: W1[n, h=col, k] ----
      const float* wr = &W1L[n * 66 + hc * 11];
      v2f B0 = { wr[0 + fb] * wmask, wr[1 + fb] * wmask };
      v2f B1 = { wr[4 + fb] * wmask, wr[5 + fb] * wmask };
      v2f B2 = { (hi ? wr[10] : wr[8]) * wmask, wr[9] * ymask };  // unconditional load
      // ---- C preloaded with broadcast bias (b1P is zero-padded, no masking) ----
      const float cb = b1P[n * 16 + m];
      v8f C = {cb, cb, cb, cb, cb, cb, cb, cb};
      C = __builtin_amdgcn_wmma_f32_16x16x4_f32(false, A0, false, B0, (short)0, C, false, false);
      C = __builtin_amdgcn_wmma_f32_16x16x4_f32(false, A1, false, B1, (short)0, C, false, false);
      C = __builtin_amdgcn_wmma_f32_16x16x4_f32(false, A2, false, B2, (short)0, C, false, false);

      // ---- transpose D (lane=col h, reg=row m) through LDS ----
      const int srb = hi ? 8 : 0;
#pragma unroll
      for (int r = 0; r < 8; ++r) hb[(srb + r) * HSTR + m] = C[r];
      asm volatile("s_wait_dscnt 0" ::: "memory");

      // ---- layer 2: lane = sample ----
      const float4 h0 = *(const float4*)(&hb[m * HSTR + 0]);
      const float4 h1 = *(const float4*)(&hb[m * HSTR + 4]);
      const float4 w0 = *(const float4*)(&W2P[n * 8 + 0]);
      const float4 w1 = *(const float4*)(&W2P[n * 8 + 4]);   // .z/.w are zero
      float y = b2L[n];
      y += fmaxf(h0.x, 0.0f) * w0.x + fmaxf(h0.y, 0.0f) * w0.y
         + fmaxf(h0.z, 0.0f) * w0.z + fmaxf(h0.w, 0.0f) * w0.w;
      y += fmaxf(h1.x, 0.0f) * w1.x + fmaxf(h1.y, 0.0f) * w1.y;
      const float o = -12.0f + 24.0f * fast_rcp(1.0f + __expf(-0.2f * y));
      if (!hi) Yl[(sw0 + m) * YSTR + n] = o;
    }
    __syncthreads();   // all reads of this buffer done before it is re-staged
  }

  // coalesced output: rows of 64 floats per sample
  for (int i = tid; i < SAMPS * 64; i += TPB) {
    int s = i >> 6, n = i & 63;
    out[(b0 + s) * 64 + n] = Yl[s * YSTR + n];
  }
}

extern "C" void kernel_launch(void* const* d_in, const int* in_sizes, int n_in,
                              void* d_out, int out_size, void* d_ws, size_t ws_size,
                              hipStream_t stream) {
  const float* X  = (const float*)d_in[0];
  const float* Z  = (const float*)d_in[1];
  const float* W1 = (const float*)d_in[2];
  const float* b1 = (const float*)d_in[3];
  const float* W2 = (const float*)d_in[4];
  const float* b2 = (const float*)d_in[5];
  float* out = (float*)d_out;
  float* ws  = (float*)d_ws;

  // ws layout: [0..2*COLS) = mu|inv_sd, then nb*2*COLS partials
  long avail = (long)(ws_size / 4) - 2 * COLS;
  int nb = (int)(avail / (2 * COLS));
  if (nb < 1)   nb = 1;
  if (nb > 256) nb = 256;
  int chunk = (BTOT + nb - 1) / nb;

  k_stats_partial<<<nb, COLS, 0, stream>>>(X, ws + 2 * COLS, nb, chunk);
  k_stats_final<<<1, COLS, 0, stream>>>(ws, nb);
  k_mlp<<<BTOT / SAMPS, TPB, 0, stream>>>(X, Z, W1, b1, W2, b2, ws, out);
}